// CustomLoss_51677046505531
// MI455X (gfx1250) — compile-verified
//
#include <hip/hip_runtime.h>
#include <hip/hip_bf16.h>

typedef float v2f __attribute__((ext_vector_type(2)));
typedef float v8f __attribute__((ext_vector_type(8)));

#define LOSS_ALPHA 0.5f
#define N 8192
#define TILE 256                      // block super-tile edge (i and j)
#define NTB  (N / TILE)               // 32 tiles per dimension
#define NBLK (NTB * NTB)              // 1024 blocks
#define NWAVE 8                       // waves per block (256 threads, wave32)

// Pairwise ranking hinge over one 256x256 (i,j) super-tile.
// Each wave: 2 strips of 16 i-rows, sweeping 16 sub-tiles of 16 j-cols.
// Per 16x16 tile: T[m][n] = 1 - x_i[m] + x_j[n] built by one
// V_WMMA_F32_16X16X4_F32 (rank-2 + constant as a K=4 matmul).
// VALU per element: cmp(label), cmp(T>0), cndmask, add. Pair counting is
// done on the SCALAR pipe: ballot reuses the label v_cmp's SGPR mask and
// popcount lowers to s_bcnt1 + s_add (co-issues with VALU, 0 VALU slots).
__global__ __launch_bounds__(256) void rank_tiles_kernel(
    const float* __restrict__ logits, const float* __restrict__ labels,
    float* __restrict__ wsHinge, unsigned int* __restrict__ wsCnt)
{
    __shared__ float xi[TILE], li[TILE], xj[TILE], lj[TILE];
    __shared__ float redH[256];
    __shared__ unsigned int redC[NWAVE];

    const int tid = threadIdx.x;
    const int bi = blockIdx.x / NTB;     // i super-tile
    const int bj = blockIdx.x % NTB;     // j super-tile

    {   // stage both tiles in LDS (1 element per thread per array)
        const int gi = bi * TILE + tid;
        const int gj = bj * TILE + tid;
        xi[tid] = logits[gi];  li[tid] = labels[gi];
        xj[tid] = logits[gj];  lj[tid] = labels[gj];
    }
    __syncthreads();

    const int wid  = tid >> 5;           // wave 0..7
    const int lane = tid & 31;
    const int half = lane >> 4;          // 0: lanes 0-15, 1: lanes 16-31
    const int col  = lane & 15;          // D column N = lane%16

    float hinge0 = 0.0f, hinge1 = 0.0f;  // two chains -> looser dependences
    unsigned int cnt = 0;                // wave-uniform, lives in SGPR

    #pragma unroll
    for (int rr = 0; rr < 2; ++rr) {
        const int r = wid * 2 + rr;      // i sub-row strip (16 i values)

        // A (16x4): col K0 = 1, K1 = x_i[m], K2 = 1, K3 = 0
        // layout: VGPR0 lanes0-15 = A[m][0], lanes16-31 = A[m][2]
        //         VGPR1 lanes0-15 = A[m][1], lanes16-31 = A[m][3]
        v2f a;
        a.x = 1.0f;
        a.y = half ? 0.0f : xi[r * 16 + col];

        // labels[i] for this lane's 8 D rows: m = half*8 + k
        float labi[8];
        #pragma unroll
        for (int k = 0; k < 8; ++k) labi[k] = li[r * 16 + half * 8 + k];

        for (int c = 0; c < TILE / 16; ++c) {
            // B (4x16): row0 = x_j[n], row1 = -1, row2 = 1, row3 = 0
            // layout: VGPR0 lanes0-15 = B[0][n], lanes16-31 = B[2][n]
            //         VGPR1 lanes0-15 = B[1][n], lanes16-31 = B[3][n]
            v2f b;
            b.x = half ? 1.0f : xj[c * 16 + col];
            b.y = half ? 0.0f : -1.0f;
            const float labj = lj[c * 16 + col];

            v8f d = {};
            // T = A x B + 0  ->  v_wmma_f32_16x16x4_f32
            d = __builtin_amdgcn_wmma_f32_16x16x4_f32(
                    false, a, false, b, (short)0, d, false, false);

            #pragma unroll
            for (int k = 0; k < 8; ++k) {
                const bool  sel = labi[k] > labj;   // labels[i] > labels[j]
                const float dk  = d[k];
                // count on the scalar pipe: ballot == the v_cmp SGPR mask,
                // popcount -> s_bcnt1_i32_b32 + s_add (zero VALU slots)
                cnt += (unsigned int)__builtin_popcount(
                           __builtin_amdgcn_ballot_w32(sel));
                // relu folded into the select: 2x v_cmp -> s_and -> cndmask -> add
                const float hv = (sel && (dk > 0.0f)) ? dk : 0.0f;
                if (k & 1) hinge1 += hv; else hinge0 += hv;
            }
        }
    }

    const float hinge = hinge0 + hinge1;

    // deterministic block tree reduction (hinge per-thread, count per-wave)
    redH[tid] = hinge;
    if (lane == 0) redC[wid] = cnt;      // cnt is wave-uniform
    __syncthreads();
    for (int s = 128; s > 0; s >>= 1) {
        if (tid < s) redH[tid] += redH[tid + s];
        __syncthreads();
    }
    if (tid == 0) {
        unsigned int ctot = 0;
        #pragma unroll
        for (int w = 0; w < NWAVE; ++w) ctot += redC[w];
        wsHinge[blockIdx.x] = redH[0];
        wsCnt[blockIdx.x]   = ctot;
    }
}

// Single-block finalize: MSE + fixed-order reduction of block partials.
__global__ __launch_bounds__(256) void finalize_kernel(
    const float* __restrict__ logits, const float* __restrict__ labels,
    const float* __restrict__ wsHinge, const unsigned int* __restrict__ wsCnt,
    float* __restrict__ out)
{
    __shared__ float rm[256], rh[256];
    __shared__ unsigned int rc[256];
    const int tid = threadIdx.x;

    float mse = 0.0f;
    for (int i = tid; i < N; i += 256) {
        const float d = logits[i] - labels[i];
        mse += d * d;
    }
    float h = 0.0f;
    unsigned int c = 0;
    for (int b = tid; b < NBLK; b += 256) { h += wsHinge[b]; c += wsCnt[b]; }

    rm[tid] = mse;  rh[tid] = h;  rc[tid] = c;
    __syncthreads();
    for (int s = 128; s > 0; s >>= 1) {
        if (tid < s) {
            rm[tid] += rm[tid + s];
            rh[tid] += rh[tid + s];
            rc[tid] += rc[tid + s];
        }
        __syncthreads();
    }
    if (tid == 0) {
        const float msev    = rm[0] / (float)N;
        const float ranking = (rc[0] > 0u) ? (rh[0] / (float)rc[0]) : 0.0f;
        out[0] = LOSS_ALPHA * msev + (1.0f - LOSS_ALPHA) * ranking;
    }
}

extern "C" void kernel_launch(void* const* d_in, const int* in_sizes, int n_in,
                              void* d_out, int out_size, void* d_ws, size_t ws_size,
                              hipStream_t stream) {
    (void)in_sizes; (void)n_in; (void)out_size; (void)ws_size;
    const float* logits = (const float*)d_in[0];
    const float* labels = (const float*)d_in[1];
    float* outp = (float*)d_out;

    float*        wsHinge = (float*)d_ws;                    // NBLK floats
    unsigned int* wsCnt   = (unsigned int*)(wsHinge + NBLK); // NBLK u32 (4 KB)

    rank_tiles_kernel<<<NBLK, 256, 0, stream>>>(logits, labels, wsHinge, wsCnt);
    finalize_kernel<<<1, 256, 0, stream>>>(logits, labels, wsHinge, wsCnt, outp);
}